// M_MambaAtten_39822936768516
// MI455X (gfx1250) — compile-verified
//
#include <hip/hip_runtime.h>
#include <hip/hip_bf16.h>
#include <math.h>

#define BATCH 4
#define SEQ   2048
#define DIMX  192
#define DI    384
#define NST   16
#define DTR   12
#define XZW   768           // 2*DI, row stride of xz buffer
#define DBLW  48            // DTR + 2*NST (=44) padded to 48
#define MROWS (BATCH*SEQ)   // 8192
#define NCH   32            // scan chunks per sequence
#define CL    (SEQ/NCH)     // 64 timesteps per chunk

typedef float v2f __attribute__((ext_vector_type(2)));
typedef float v8f __attribute__((ext_vector_type(8)));

// Native fp32 WMMA: D(16x16,f32) = A(16x4,f32) * B(4x16,f32) + C
__device__ __forceinline__ v8f wmma4(v2f a, v2f b, v8f c) {
  return __builtin_amdgcn_wmma_f32_16x16x4_f32(false, a, false, b, (short)0, c,
                                               false, false);
}

// ---------------------------------------------------------------------------
// xz = x @ W_in^T  (M=8192, K=192, N=768). One wave computes a 16x64 strip
// (4 N-tiles): A fragment loaded once, reused by 4 WMMAs.
// ---------------------------------------------------------------------------
__global__ void k_gemm_in(const float* __restrict__ x,
                          const float* __restrict__ Win,
                          float* __restrict__ xz, int reverse) {
  const int wave = blockIdx.x * (blockDim.x >> 5) + (threadIdx.x >> 5);
  const int lane = threadIdx.x & 31;
  const int TG = XZW / 64;                 // 12 N-groups of 64
  const int gm = wave / TG, gng = wave % TG;
  const int mrow = lane & 15, kh = lane >> 4;

  int m  = gm * 16 + mrow;
  int bb = m / SEQ, ll = m % SEQ;
  int srow = reverse ? (bb * SEQ + (SEQ - 1 - ll)) : m;

  const float* ap = x + (size_t)srow * DIMX;
  const float* bp[4];
#pragma unroll
  for (int j = 0; j < 4; ++j)
    bp[j] = Win + (size_t)(gng * 64 + j * 16 + mrow) * DIMX;

  v8f acc[4] = {};
  for (int k0 = 0; k0 < DIMX; k0 += 4) {
    int kk = k0 + 2 * kh;
    v2f a; a.x = ap[kk]; a.y = ap[kk + 1];
#pragma unroll
    for (int j = 0; j < 4; ++j) {
      v2f b; b.x = bp[j][kk]; b.y = bp[j][kk + 1];
      acc[j] = wmma4(a, b, acc[j]);
    }
  }
#pragma unroll
  for (int j = 0; j < 4; ++j)
#pragma unroll
    for (int v = 0; v < 8; ++v) {
      int orow = gm * 16 + v + 8 * kh;
      xz[(size_t)orow * XZW + gng * 64 + j * 16 + mrow] = acc[j][v];
    }
}

// ---------------------------------------------------------------------------
// depthwise causal conv1d (taps=4) + SiLU:  u = silu(conv(xc) + bias)
// ---------------------------------------------------------------------------
__global__ void k_conv_silu(const float* __restrict__ xz,
                            const float* __restrict__ cw,
                            const float* __restrict__ cb,
                            float* __restrict__ u) {
  int idx = blockIdx.x * blockDim.x + threadIdx.x;
  if (idx >= MROWS * DI) return;
  int d = idx % DI;
  int t = (idx / DI) % SEQ;
  int b = idx / (DI * SEQ);
  float acc = cb[d];
#pragma unroll
  for (int j = 0; j < 4; ++j) {
    int tt = t - 3 + j;
    if (tt >= 0)
      acc += xz[(size_t)(b * SEQ + tt) * XZW + d] * cw[d * 4 + j];
  }
  u[(size_t)(b * SEQ + t) * DI + d] = acc / (1.0f + __expf(-acc));
}

// ---------------------------------------------------------------------------
// dbl = u @ W_xp^T  (M=8192, K=384, N=44 pad 48). One wave: all 3 N-tiles.
// ---------------------------------------------------------------------------
__global__ void k_gemm_xp(const float* __restrict__ u,
                          const float* __restrict__ Wxp,
                          float* __restrict__ dbl) {
  const int wave = blockIdx.x * (blockDim.x >> 5) + (threadIdx.x >> 5);
  const int lane = threadIdx.x & 31;
  const int gm = wave;                     // 512 M-tiles
  const int mrow = lane & 15, kh = lane >> 4;

  const float* ap = u + (size_t)(gm * 16 + mrow) * DI;
  bool nvalid[3];
  const float* bp[3];
#pragma unroll
  for (int j = 0; j < 3; ++j) {
    int n = j * 16 + mrow;
    nvalid[j] = (n < DTR + 2 * NST);       // n < 44
    bp[j] = Wxp + (size_t)(nvalid[j] ? n : 0) * DI;
  }

  v8f acc[3] = {};
  for (int k0 = 0; k0 < DI; k0 += 4) {
    int kk = k0 + 2 * kh;
    v2f a; a.x = ap[kk]; a.y = ap[kk + 1];
#pragma unroll
    for (int j = 0; j < 3; ++j) {
      v2f b;
      b.x = nvalid[j] ? bp[j][kk] : 0.0f;
      b.y = nvalid[j] ? bp[j][kk + 1] : 0.0f;
      acc[j] = wmma4(a, b, acc[j]);
    }
  }
#pragma unroll
  for (int j = 0; j < 3; ++j)
#pragma unroll
    for (int v = 0; v < 8; ++v) {
      int orow = gm * 16 + v + 8 * kh;
      dbl[(size_t)orow * DBLW + j * 16 + mrow] = acc[j][v];
    }
}

// ---------------------------------------------------------------------------
// dt = softplus(dt_raw @ W_dt^T + b_dt)  (M=8192, K=12, N=384)
// 4 N-tiles per wave; result overwrites xc region (cols [0,DI)) of xz.
// ---------------------------------------------------------------------------
__global__ void k_gemm_dt(const float* __restrict__ dbl,
                          const float* __restrict__ Wdt,
                          const float* __restrict__ bdt,
                          float* __restrict__ xz) {
  const int wave = blockIdx.x * (blockDim.x >> 5) + (threadIdx.x >> 5);
  const int lane = threadIdx.x & 31;
  const int TG = DI / 64;                  // 6 N-groups
  const int gm = wave / TG, gng = wave % TG;
  const int mrow = lane & 15, kh = lane >> 4;

  const float* ap = dbl + (size_t)(gm * 16 + mrow) * DBLW;
  const float* bp[4];
#pragma unroll
  for (int j = 0; j < 4; ++j)
    bp[j] = Wdt + (size_t)(gng * 64 + j * 16 + mrow) * DTR;

  v8f acc[4] = {};
  for (int k0 = 0; k0 < DTR; k0 += 4) {    // 3 iters, k <= 11
    int kk = k0 + 2 * kh;
    v2f a; a.x = ap[kk]; a.y = ap[kk + 1];
#pragma unroll
    for (int j = 0; j < 4; ++j) {
      v2f b; b.x = bp[j][kk]; b.y = bp[j][kk + 1];
      acc[j] = wmma4(a, b, acc[j]);
    }
  }
#pragma unroll
  for (int j = 0; j < 4; ++j) {
    int n = gng * 64 + j * 16 + mrow;
    float bias = bdt[n];
#pragma unroll
    for (int v = 0; v < 8; ++v) {
      int orow = gm * 16 + v + 8 * kh;
      float t = acc[j][v] + bias;
      float sp = (t > 20.0f) ? t : log1pf(__expf(t));    // softplus
      xz[(size_t)orow * XZW + n] = sp;
    }
  }
}

// ---------------------------------------------------------------------------
// Chunked selective scan, pass 1: per chunk compute P[n]=prod(dA) and local
// end-state S[n] (h_init = 0). Blocks = BATCH*NCH, threads = DI channels.
// ---------------------------------------------------------------------------
__global__ void k_scan_part1(const float* __restrict__ xz,
                             const float* __restrict__ u,
                             const float* __restrict__ dbl,
                             const float* __restrict__ Alog,
                             float* __restrict__ Pb, float* __restrict__ Sb) {
  const int b = blockIdx.x / NCH;
  const int c = blockIdx.x % NCH;
  const int d = threadIdx.x;

  float a[NST], P[NST], S[NST];
#pragma unroll
  for (int n = 0; n < NST; ++n) {
    a[n] = -__expf(Alog[d * NST + n]);
    P[n] = 1.0f;
    S[n] = 0.0f;
  }
  const int t0 = c * CL;
  for (int t = t0; t < t0 + CL; ++t) {
    const size_t row = (size_t)b * SEQ + t;
    const float* bc = dbl + row * DBLW;
    __builtin_prefetch(bc + DBLW, 0, 0);                 // next timestep B/C
    const float dt = xz[row * XZW + d];
    const float ut = u[row * DI + d];
    const float dtu = dt * ut;
#pragma unroll
    for (int n = 0; n < NST; ++n) {
      float dA = __expf(dt * a[n]);
      S[n] = S[n] * dA + dtu * bc[DTR + n];
      P[n] *= dA;
    }
  }
  const size_t base = ((size_t)(b * NCH + c) * DI + d) * NST;
#pragma unroll
  for (int n = 0; n < NST; ++n) {
    Pb[base + n] = P[n];
    Sb[base + n] = S[n];
  }
}

// ---------------------------------------------------------------------------
// Pass 2: sequential combine over NCH chunks, parallel over (b,d,n).
// In place: Sb[c] is rewritten with h_init for chunk c.
// ---------------------------------------------------------------------------
__global__ void k_scan_combine(const float* __restrict__ Pb,
                               float* __restrict__ Sb) {
  int idx = blockIdx.x * blockDim.x + threadIdx.x;
  if (idx >= BATCH * DI * NST) return;
  const int n = idx & (NST - 1);
  const int d = (idx >> 4) % DI;
  const int b = idx / (NST * DI);

  float carry = 0.0f;
  for (int c = 0; c < NCH; ++c) {
    const size_t base = ((size_t)(b * NCH + c) * DI + d) * NST + n;
    float p = Pb[base];
    float s = Sb[base];
    Sb[base] = carry;                      // h_init for chunk c
    carry = p * carry + s;
  }
}

// ---------------------------------------------------------------------------
// Pass 3: re-run each chunk with corrected h_init, emit y, fuse +u*D and
// silu(z) gate; gated y overwrites z region of xz in place.
// ---------------------------------------------------------------------------
__global__ void k_scan_part3(float* __restrict__ xz,
                             const float* __restrict__ u,
                             const float* __restrict__ dbl,
                             const float* __restrict__ Alog,
                             const float* __restrict__ Dv,
                             const float* __restrict__ Sb) {
  const int b = blockIdx.x / NCH;
  const int c = blockIdx.x % NCH;
  const int d = threadIdx.x;

  float a[NST], h[NST];
  const size_t base = ((size_t)(b * NCH + c) * DI + d) * NST;
#pragma unroll
  for (int n = 0; n < NST; ++n) {
    a[n] = -__expf(Alog[d * NST + n]);
    h[n] = Sb[base + n];                   // corrected initial state
  }
  const float Dd = Dv[d];

  const int t0 = c * CL;
  for (int t = t0; t < t0 + CL; ++t) {
    const size_t row = (size_t)b * SEQ + t;
    const float* bc = dbl + row * DBLW;
    __builtin_prefetch(bc + DBLW, 0, 0);
    const float dt = xz[row * XZW + d];
    const float ut = u[row * DI + d];
    const float dtu = dt * ut;
    float y = 0.0f;
#pragma unroll
    for (int n = 0; n < NST; ++n) {
      float dA = __expf(dt * a[n]);
      h[n] = h[n] * dA + dtu * bc[DTR + n];          // B_t
      y += h[n] * bc[DTR + NST + n];                 // C_t
    }
    y += ut * Dd;
    float z = xz[row * XZW + DI + d];
    y *= z / (1.0f + __expf(-z));                    // * silu(z)
    xz[row * XZW + DI + d] = y;                      // overwrite z
  }
}

// ---------------------------------------------------------------------------
// out = x + (y_f + reverse(y_b)) @ W_out^T  (M=8192, K=384, N=192)
// 4 N-tiles per wave; fused bidirectional merge on A load, residual epilogue.
// ---------------------------------------------------------------------------
__global__ void k_gemm_out(const float* __restrict__ xz_f,
                           const float* __restrict__ xz_b,
                           const float* __restrict__ x,
                           const float* __restrict__ Wout,
                           float* __restrict__ out) {
  const int wave = blockIdx.x * (blockDim.x >> 5) + (threadIdx.x >> 5);
  const int lane = threadIdx.x & 31;
  const int TG = DIMX / 64;                // 3 N-groups
  const int gm = wave / TG, gng = wave % TG;
  const int mrow = lane & 15, kh = lane >> 4;

  int m  = gm * 16 + mrow;
  int bb = m / SEQ, ll = m % SEQ;
  int mrev = bb * SEQ + (SEQ - 1 - ll);

  const float* af = xz_f + (size_t)m    * XZW + DI;
  const float* ab = xz_b + (size_t)mrev * XZW + DI;
  const float* bp[4];
#pragma unroll
  for (int j = 0; j < 4; ++j)
    bp[j] = Wout + (size_t)(gng * 64 + j * 16 + mrow) * DI;

  v8f acc[4] = {};
  for (int k0 = 0; k0 < DI; k0 += 4) {
    int kk = k0 + 2 * kh;
    v2f a; a.x = af[kk] + ab[kk]; a.y = af[kk + 1] + ab[kk + 1];
#pragma unroll
    for (int j = 0; j < 4; ++j) {
      v2f b; b.x = bp[j][kk]; b.y = bp[j][kk + 1];
      acc[j] = wmma4(a, b, acc[j]);
    }
  }
#pragma unroll
  for (int j = 0; j < 4; ++j)
#pragma unroll
    for (int v = 0; v < 8; ++v) {
      int orow = gm * 16 + v + 8 * kh;
      int n = gng * 64 + j * 16 + mrow;
      out[(size_t)orow * DIMX + n] = acc[j][v] + x[(size_t)orow * DIMX + n];
    }
}

// ---------------------------------------------------------------------------
extern "C" void kernel_launch(void* const* d_in, const int* in_sizes, int n_in,
                              void* d_out, int out_size, void* d_ws, size_t ws_size,
                              hipStream_t stream) {
  const float* x      = (const float*)d_in[0];
  const float* Win_f  = (const float*)d_in[1];
  const float* cw_f   = (const float*)d_in[2];
  const float* cb_f   = (const float*)d_in[3];
  const float* Wxp_f  = (const float*)d_in[4];
  const float* Wdt_f  = (const float*)d_in[5];
  const float* bdt_f  = (const float*)d_in[6];
  const float* Alog_f = (const float*)d_in[7];
  const float* Dv_f   = (const float*)d_in[8];
  const float* Win_b  = (const float*)d_in[9];
  const float* cw_b   = (const float*)d_in[10];
  const float* cb_b   = (const float*)d_in[11];
  const float* Wxp_b  = (const float*)d_in[12];
  const float* Wdt_b  = (const float*)d_in[13];
  const float* bdt_b  = (const float*)d_in[14];
  const float* Alog_b = (const float*)d_in[15];
  const float* Dv_b   = (const float*)d_in[16];
  const float* Wout   = (const float*)d_in[17];
  float* out = (float*)d_out;

  // workspace (floats): 2*6291456 + 2*3145728 + 2*393216 + 4*786432 ≈ 91 MB
  float* ws    = (float*)d_ws;
  float* xz_f  = ws;                                   // 8192*768
  float* xz_b  = xz_f + (size_t)MROWS * XZW;
  float* u_f   = xz_b + (size_t)MROWS * XZW;           // 8192*384
  float* u_b   = u_f  + (size_t)MROWS * DI;
  float* dbl_f = u_b  + (size_t)MROWS * DI;            // 8192*48
  float* dbl_b = dbl_f + (size_t)MROWS * DBLW;
  const size_t PS = (size_t)BATCH * NCH * DI * NST;    // 786432
  float* P_f   = dbl_b + (size_t)MROWS * DBLW;
  float* S_f   = P_f + PS;
  float* P_b   = S_f + PS;
  float* S_b   = P_b + PS;

  const int BLK = 256, WPB = BLK / 32;

  // 1) input projections (fwd + reversed)
  {
    int waves = (MROWS / 16) * (XZW / 64);             // 6144
    k_gemm_in<<<waves / WPB, BLK, 0, stream>>>(x, Win_f, xz_f, 0);
    k_gemm_in<<<waves / WPB, BLK, 0, stream>>>(x, Win_b, xz_b, 1);
  }
  // 2) depthwise conv + silu -> u
  {
    int n = MROWS * DI;
    k_conv_silu<<<(n + BLK - 1) / BLK, BLK, 0, stream>>>(xz_f, cw_f, cb_f, u_f);
    k_conv_silu<<<(n + BLK - 1) / BLK, BLK, 0, stream>>>(xz_b, cw_b, cb_b, u_b);
  }
  // 3) x_dbl = u @ W_xp^T
  {
    int waves = MROWS / 16;                            // 512
    k_gemm_xp<<<waves / WPB, BLK, 0, stream>>>(u_f, Wxp_f, dbl_f);
    k_gemm_xp<<<waves / WPB, BLK, 0, stream>>>(u_b, Wxp_b, dbl_b);
  }
  // 4) dt = softplus(...) -> xc region of xz
  {
    int waves = (MROWS / 16) * (DI / 64);              // 3072
    k_gemm_dt<<<waves / WPB, BLK, 0, stream>>>(dbl_f, Wdt_f, bdt_f, xz_f);
    k_gemm_dt<<<waves / WPB, BLK, 0, stream>>>(dbl_b, Wdt_b, bdt_b, xz_b);
  }
  // 5) chunk-parallel selective scan (3 passes, both branches)
  {
    int cb = (BATCH * DI * NST + BLK - 1) / BLK;       // combine blocks
    k_scan_part1<<<BATCH * NCH, DI, 0, stream>>>(xz_f, u_f, dbl_f, Alog_f, P_f, S_f);
    k_scan_part1<<<BATCH * NCH, DI, 0, stream>>>(xz_b, u_b, dbl_b, Alog_b, P_b, S_b);
    k_scan_combine<<<cb, BLK, 0, stream>>>(P_f, S_f);
    k_scan_combine<<<cb, BLK, 0, stream>>>(P_b, S_b);
    k_scan_part3<<<BATCH * NCH, DI, 0, stream>>>(xz_f, u_f, dbl_f, Alog_f, Dv_f, S_f);
    k_scan_part3<<<BATCH * NCH, DI, 0, stream>>>(xz_b, u_b, dbl_b, Alog_b, Dv_b, S_b);
  }
  // 6) out-proj with fused bidirectional merge + residual
  {
    int waves = (MROWS / 16) * (DIMX / 64);            // 1536
    k_gemm_out<<<waves / WPB, BLK, 0, stream>>>(xz_f, xz_b, x, Wout, out);
  }
}